// GraphDecoderUnit_13692355739967
// MI455X (gfx1250) — compile-verified
//
#include <hip/hip_runtime.h>

typedef __attribute__((ext_vector_type(2))) float v2f;
typedef __attribute__((ext_vector_type(8))) float v8f;

#define LDS_STRIDE 260   // 260 mod 64 = 4 -> conflict-free WMMA fragment reads

// ---------------------------------------------------------------------------
// Kernel 0: zero a buffer (float4 grid-stride; all our sizes are %4 == 0)
// ---------------------------------------------------------------------------
__global__ void zero_kernel(float* __restrict__ p, long n) {
    long i = ((long)blockIdx.x * blockDim.x + threadIdx.x) * 4;
    const long stride = (long)gridDim.x * blockDim.x * 4;
    for (; i + 3 < n; i += stride) {
        *(float4*)(p + i) = make_float4(0.f, 0.f, 0.f, 0.f);
    }
}

// ---------------------------------------------------------------------------
// Kernel 1: gather + segment-sum.  32 threads per edge, 4 channels each.
// agg[c, r[e]] += x[c, g[e]],  x = concat(a, b) over 128 channels.
// x and agg are L2-resident (64 MB each), so random columns hit L2.
// Pointer-select (no divergent branch) between a and b.
// ---------------------------------------------------------------------------
__global__ void seg_sum_kernel(const float* __restrict__ a,
                               const float* __restrict__ b,
                               const int*   __restrict__ gidx,
                               const int*   __restrict__ ridx,
                               float*       __restrict__ agg,
                               long N, long E) {
    long t = (long)blockIdx.x * blockDim.x + threadIdx.x;
    long e = t >> 5;
    if (e >= E) return;
    int  c4 = (int)(t & 31) * 4;
    long g  = gidx[e];
    long r  = ridx[e];
#pragma unroll
    for (int j = 0; j < 4; ++j) {
        int   c    = c4 + j;
        const float* base = (c < 64) ? a : b;        // cndmask, not a branch
        float v = base[(long)(c & 63) * N + g];
        atomicAdd(&agg[(long)c * N + r], v);
    }
}

// ---------------------------------------------------------------------------
// Kernel 2: h = relu([W_self | W_neigh] @ [x; agg] + bias) via
// V_WMMA_F32_16X16X4_F32, operands staged through LDS so the inner loop is
// branch-free.  256 threads = 8 waves; wave (mtile, nsub) owns one 16x16 tile
// of the block's 64x32 output tile.  K = 256 accumulated in steps of 4.
// ---------------------------------------------------------------------------
__global__ void __launch_bounds__(256)
gemm_wmma_kernel(const float* __restrict__ a,
                 const float* __restrict__ b,
                 const float* __restrict__ Ws,
                 const float* __restrict__ Wn,
                 const float* __restrict__ bias,
                 const float* __restrict__ agg,
                 float*       __restrict__ h,
                 long N) {
    __shared__ float Wc[64 * LDS_STRIDE];   // Wcat[row][k], row-major, padded
    __shared__ float Xs[32 * LDS_STRIDE];   // Xs[col][k], col-major, padded

    const long nbase = (long)blockIdx.x * 32;

    // --- Stage Wcat = [W_self | W_neigh] (64 x 256).  Thread t owns column
    // k = t for every row, so the Ws/Wn split is wave-uniform and the global
    // reads are coalesced along rows.
    {
        const int k = threadIdx.x;                     // 0..255
        const float* wsrc = (k < 128) ? (Ws + k) : (Wn + (k - 128));
        for (int r = 0; r < 64; ++r) {
            Wc[r * LDS_STRIDE + k] = wsrc[r * 128];
        }
    }

    // --- Stage X slab (256 rows x 32 cols) column-major into Xs.
    // Wave w handles rows k = w, w+8, ... (k uniform per wave -> uniform
    // branch among a/b/agg); lanes cover the 32 columns (coalesced 128B).
    {
        const int  tcol = threadIdx.x & 31;
        const int  krow0 = threadIdx.x >> 5;           // 0..7
        const long gcol = nbase + tcol;
        for (int k = krow0; k < 256; k += 8) {
            const float* src;
            if (k < 64)       src = a   + (long)k * N;
            else if (k < 128) src = b   + (long)(k - 64) * N;
            else              src = agg + (long)(k - 128) * N;
            Xs[tcol * LDS_STRIDE + k] = src[gcol];
        }
    }
    __syncthreads();

    const int lane  = threadIdx.x & 31;
    const int wid   = threadIdx.x >> 5;
    const int mtile = wid & 3;            // 4 M-tiles cover 64 output rows
    const int nsub  = wid >> 2;           // 2 N-subtiles per block
    const int l     = lane & 15;
    const int half  = lane >> 4;          // lanes 0-15: K+0/1, 16-31: K+2/3

    // Per-lane fragment base pointers (fixed across the K loop)
    const float* Arow = &Wc[(mtile * 16 + l) * LDS_STRIDE + 2 * half];
    const float* Bcol = &Xs[(nsub  * 16 + l) * LDS_STRIDE + 2 * half];

    v8f acc = {};
#pragma unroll
    for (int k0 = 0; k0 < 256; k0 += 4) {
        v2f A = *(const v2f*)(Arow + k0);   // ds_load_b64: W[row, ka..ka+1]
        v2f B = *(const v2f*)(Bcol + k0);   // ds_load_b64: X[ka..ka+1, col]
        acc = __builtin_amdgcn_wmma_f32_16x16x4_f32(
                  /*neg_a=*/false, A, /*neg_b=*/false, B,
                  /*c_mod=*/(short)0, acc, /*reuse_a=*/false, /*reuse_b=*/false);
    }

    // D layout: VGPR j -> M = mtile*16 + j + 8*half, N = nbase + nsub*16 + l
#pragma unroll
    for (int j = 0; j < 8; ++j) {
        const int m = mtile * 16 + j + 8 * half;
        float v = acc[j] + bias[m];
        v = v > 0.f ? v : 0.f;            // fused ReLU
        h[(long)m * N + nbase + nsub * 16 + l] = v;
    }
}

// ---------------------------------------------------------------------------
// Kernel 3: scatter-max unpool.  16 threads per entry, 4 channels each.
// h >= 0 (post-ReLU) and out starts at 0, so float max == u32 max on bits.
// ---------------------------------------------------------------------------
__global__ void unpool_max_kernel(const float* __restrict__ h,
                                  const int*   __restrict__ src,
                                  const int*   __restrict__ dst,
                                  float*       __restrict__ out,
                                  long N, long K, long NF) {
    long t = (long)blockIdx.x * blockDim.x + threadIdx.x;
    long k = t >> 4;
    if (k >= K) return;
    int  c4 = (int)(t & 15) * 4;
    long s  = src[k];
    long d  = dst[k];
#pragma unroll
    for (int j = 0; j < 4; ++j) {
        int   c = c4 + j;
        float v = h[(long)c * N + s];
        atomicMax((unsigned int*)&out[(long)c * NF + d], __float_as_uint(v));
    }
}

// ---------------------------------------------------------------------------
extern "C" void kernel_launch(void* const* d_in, const int* in_sizes, int n_in,
                              void* d_out, int out_size, void* d_ws, size_t ws_size,
                              hipStream_t stream) {
    const float* a    = (const float*)d_in[0];  // (64, N)
    const float* b    = (const float*)d_in[1];  // (64, N)
    const float* Ws   = (const float*)d_in[2];  // (64, 128)
    const float* Wn   = (const float*)d_in[3];  // (64, 128)
    const float* bias = (const float*)d_in[4];  // (64,)
    const int*   gidx = (const int*)d_in[5];    // (E,)
    const int*   ridx = (const int*)d_in[6];    // (E,)
    const int*   usrc = (const int*)d_in[7];    // (K,)
    const int*   udst = (const int*)d_in[8];    // (K,)
    float*       out  = (float*)d_out;          // (64, 4N)

    const long N  = (long)in_sizes[0] / 64;     // 131072
    const long E  = (long)in_sizes[5];          // 1048576
    const long K  = (long)in_sizes[7];          // 1572864
    const long NF = 4 * N;

    float* agg = (float*)d_ws;                  // 128*N floats (64 MB)
    float* h   = agg + 128 * N;                 // 64*N floats (32 MB)

    // Zero agg (atomic-add target) and out (scatter-max target).
    zero_kernel<<<2048, 256, 0, stream>>>(agg, 128 * N);
    zero_kernel<<<2048, 256, 0, stream>>>(out, (long)out_size);

    // 1) segment-sum: E edges * 32 threads
    {
        long threads = E * 32;
        int  blocks  = (int)((threads + 255) / 256);
        seg_sum_kernel<<<blocks, 256, 0, stream>>>(a, b, gidx, ridx, agg, N, E);
    }

    // 2) WMMA GEMM + bias + ReLU: each block covers a 64x32 tile of h
    {
        int blocks = (int)((N + 31) / 32);
        gemm_wmma_kernel<<<blocks, 256, 0, stream>>>(a, b, Ws, Wn, bias, agg, h, N);
    }

    // 3) scatter-max unpool: K entries * 16 threads
    {
        long threads = K * 16;
        int  blocks  = (int)((threads + 255) / 256);
        unpool_max_kernel<<<blocks, 256, 0, stream>>>(h, usrc, udst, out, N, K, NF);
    }
}